// NFFT_48679159333107
// MI455X (gfx1250) — compile-verified
//
#include <hip/hip_runtime.h>
#include <hip/hip_bf16.h>
#include <math.h>

// ---------------- NFFT constants (match reference) ----------------
#define NFULL   512           // oversampled grid n
#define NHALF   256           // output spectral size N
#define MPTS    262144        // nonequispaced points
#define MWIN    4             // window cutoff m
#define B_KB    4.7123889803846899f   // (2 - 1/sigma)*pi = 1.5*pi
#define PI_F    3.14159265358979f

typedef float v2f __attribute__((ext_vector_type(2)));
typedef float v8f __attribute__((ext_vector_type(8)));

// ---------------- I0 (modified Bessel), Abramowitz & Stegun 9.8 ----------------
__device__ __forceinline__ float bessel_i0(float t) {
    float ax = fabsf(t);
    if (ax < 3.75f) {
        float y = t / 3.75f; y *= y;
        return 1.0f + y*(3.5156229f + y*(3.0899424f + y*(1.2067492f
             + y*(0.2659732f + y*(0.0360768f + y*0.0045813f)))));
    } else {
        float u = 3.75f / ax;
        float p = 0.39894228f + u*(0.01328592f + u*(0.00225319f + u*(-0.00157565f
              + u*(0.00916281f + u*(-0.02057706f + u*(0.02635537f
              + u*(-0.01647633f + u*0.00392377f)))))));
        return (expf(ax) * rsqrtf(ax)) * p;
    }
}

// Kaiser-Bessel spatial window phi(k) with nan_to_num(->0) semantics of reference.
// nk = n*k = fx - ind  (|nk| <= MWIN)
__device__ __forceinline__ float kb_window(float nk) {
    float a2 = (float)(MWIN * MWIN) - nk * nk;
    if (a2 > 0.0f) {
        float a = sqrtf(a2);
        return sinhf(B_KB * a) / (a * PI_F);
    }
    return 0.0f;   // sqrt(neg) -> NaN -> 0, and a==0 -> NaN -> 0
}

// ---------------- Kernel 0: build twiddle matrix D and deconvolved padded Ghat ----
// D[j][k] = exp(-2*pi*i * (((j+256)%512)*((k+256)%512) mod 512) / 512)
// (folds fftshift before FFT and ifftshift after FFT into the DFT matrix; D symmetric)
__global__ __launch_bounds__(256)
void nfft_init_kernel(const float* __restrict__ fhr, const float* __restrict__ fhi,
                      float* __restrict__ Dr, float* __restrict__ Di,
                      float* __restrict__ Gr, float* __restrict__ Gi) {
    int t = blockIdx.x * blockDim.x + threadIdx.x;   // 0 .. 512*512-1 (exact)
    int i = t >> 9;
    int j = t & (NFULL - 1);

    // twiddle
    int ri = (i + NHALF) & (NFULL - 1);
    int rj = (j + NHALF) & (NFULL - 1);
    int p  = (ri * rj) & (NFULL - 1);
    float ang = -(float)p * (2.0f * PI_F / (float)NFULL);
    float s, c;
    __sincosf(ang, &s, &c);
    Dr[t] = c;
    Di[t] = s;

    // deconvolved + zero-padded g_hat (center 256x256 region)
    float gr = 0.0f, gi = 0.0f;
    if (i >= 128 && i < 384 && j >= 128 && j < 384) {
        int k0 = i - 256;                 // k in [-128,128)
        int k1 = j - 256;
        float a0 = (float)k0 * (PI_F / 256.0f);   // 2*pi*k/512
        float a1 = (float)k1 * (PI_F / 256.0f);
        float t0 = (float)MWIN * sqrtf(B_KB * B_KB - a0 * a0);
        float t1 = (float)MWIN * sqrtf(B_KB * B_KB - a1 * a1);
        float ph = bessel_i0(t0) * bessel_i0(t1);
        int src = (i - 128) * NHALF + (j - 128);
        float inv = 1.0f / ph;
        gr = fhr[src] * inv;
        gi = fhi[src] * inv;
    }
    Gr[t] = gr;
    Gi[t] = gi;
}

// ---------------- Kernel 1: complex GEMM 512x512x512 via V_WMMA_F32_16X16X4_F32 ---
// C = A * B  (complex, split real/imag). One wave -> one 16x16 tile.
// 1024 tiles total, 8 waves/block, 128 blocks => EXEC all-1s everywhere (WMMA req).
__global__ __launch_bounds__(256)
void cgemm512_wmma_kernel(const float* __restrict__ Ar, const float* __restrict__ Ai,
                          const float* __restrict__ Br, const float* __restrict__ Bi,
                          float* __restrict__ Cr, float* __restrict__ Ci,
                          float2* __restrict__ Cpack, int pack) {
    const int lane = threadIdx.x & 31;
    const int wid  = threadIdx.x >> 5;
    const int tile = blockIdx.x * 8 + wid;          // 0..1023
    const int tm   = (tile >> 5) << 4;              // tile row * 16
    const int tn   = (tile & 31) << 4;              // tile col * 16

    const int hl   = lane & 15;
    const int kh   = lane >> 4;                     // 0 or 1 (K half)
    const int mrow = tm + hl;                       // A row for this lane
    const int ncol = tn + hl;                       // B col for this lane

    v8f acc_rr = {}; v8f acc_ii = {}; v8f acc_ri = {}; v8f acc_ir = {};

    #pragma unroll 4
    for (int k = 0; k < NFULL; k += 4) {
        int ka = k + 2 * kh;
        // A (16x4): lane holds {A[m][ka], A[m][ka+1]}
        v2f ar = *(const v2f*)(Ar + mrow * NFULL + ka);
        v2f ai = *(const v2f*)(Ai + mrow * NFULL + ka);
        // B (4x16): lane holds {B[ka][n], B[ka+1][n]}
        v2f br; br.x = Br[ka * NFULL + ncol]; br.y = Br[(ka + 1) * NFULL + ncol];
        v2f bi; bi.x = Bi[ka * NFULL + ncol]; bi.y = Bi[(ka + 1) * NFULL + ncol];

        acc_rr = __builtin_amdgcn_wmma_f32_16x16x4_f32(false, ar, false, br, (short)0, acc_rr, false, false);
        acc_ii = __builtin_amdgcn_wmma_f32_16x16x4_f32(false, ai, false, bi, (short)0, acc_ii, false, false);
        acc_ri = __builtin_amdgcn_wmma_f32_16x16x4_f32(false, ar, false, bi, (short)0, acc_ri, false, false);
        acc_ir = __builtin_amdgcn_wmma_f32_16x16x4_f32(false, ai, false, br, (short)0, acc_ir, false, false);
    }

    // C/D layout: VGPR v, lanes 0-15 -> M=v, lanes 16-31 -> M=v+8; N = lane&15
    const int mbase = tm + 8 * kh;
    const int nout  = tn + hl;
    #pragma unroll
    for (int v = 0; v < 8; ++v) {
        float cr = acc_rr[v] - acc_ii[v];
        float ci = acc_ri[v] + acc_ir[v];
        int idx = (mbase + v) * NFULL + nout;
        if (pack) {
            Cpack[idx] = make_float2(cr, ci);
        } else {
            Cr[idx] = cr;
            Ci[idx] = ci;
        }
    }
}

// ---------------- Kernel 2: windowed gather (sparse convolution) -------------------
__global__ __launch_bounds__(256)
void nfft_gather_kernel(const float* __restrict__ x, const float2* __restrict__ g,
                        float* __restrict__ out) {
    int m = blockIdx.x * blockDim.x + threadIdx.x;   // 0..MPTS-1 (exact)
    float fx0 = x[2 * m]     * (float)NFULL;
    float fx1 = x[2 * m + 1] * (float)NFULL;
    int b0 = (int)ceilf(fx0) - MWIN;
    int b1 = (int)ceilf(fx1) - MWIN;

    float w0[2 * MWIN], w1[2 * MWIN];
    int   i0a[2 * MWIN], i1a[2 * MWIN];
    #pragma unroll
    for (int w = 0; w < 2 * MWIN; ++w) {
        int ind0 = b0 + w, ind1 = b1 + w;
        w0[w]  = kb_window(fx0 - (float)ind0);
        w1[w]  = kb_window(fx1 - (float)ind1);
        i0a[w] = (ind0 + NHALF + NFULL) & (NFULL - 1);   // (ind + n/2) mod n
        i1a[w] = (ind1 + NHALF + NFULL) & (NFULL - 1);
    }

    float fr = 0.0f, fi = 0.0f;
    #pragma unroll
    for (int i = 0; i < 2 * MWIN; ++i) {
        int rowoff = i0a[i] * NFULL;
        float wi = w0[i];
        #pragma unroll
        for (int j = 0; j < 2 * MWIN; ++j) {
            float2 gv = g[rowoff + i1a[j]];
            float wv = wi * w1[j];
            fr = fmaf(gv.x, wv, fr);
            fi = fmaf(gv.y, wv, fi);
        }
    }
    out[2 * m]     = fr;   // complex64 interleaved
    out[2 * m + 1] = fi;
}

// ---------------- Host-side orchestration -----------------------------------------
extern "C" void kernel_launch(void* const* d_in, const int* in_sizes, int n_in,
                              void* d_out, int out_size, void* d_ws, size_t ws_size,
                              hipStream_t stream) {
    const float* x   = (const float*)d_in[0];   // (1,1,M,2) f32
    const float* fhr = (const float*)d_in[1];   // (1,1,256,256) f32
    const float* fhi = (const float*)d_in[2];   // (1,1,256,256) f32
    float* out = (float*)d_out;                 // M complex64 -> 2*M f32 interleaved

    const size_t GRID = (size_t)NFULL * NFULL;  // 262144 elements
    float* ws = (float*)d_ws;                   // 8 * 1MB used
    float*  Dr = ws + 0 * GRID;
    float*  Di = ws + 1 * GRID;
    float*  Gr = ws + 2 * GRID;
    float*  Gi = ws + 3 * GRID;
    float*  Tr = ws + 4 * GRID;
    float*  Ti = ws + 5 * GRID;
    float2* Gp = (float2*)(ws + 6 * GRID);      // final grid, interleaved (2MB)

    // 1) twiddles + deconvolved/padded spectrum (512*512 threads, exact grid)
    nfft_init_kernel<<<GRID / 256, 256, 0, stream>>>(fhr, fhi, Dr, Di, Gr, Gi);

    // 2) T = D * Ghat   (complex GEMM, WMMA f32)
    cgemm512_wmma_kernel<<<128, 256, 0, stream>>>(Dr, Di, Gr, Gi, Tr, Ti, (float2*)nullptr, 0);

    // 3) g = T * D  (D symmetric == D^T), write interleaved for the gather
    cgemm512_wmma_kernel<<<128, 256, 0, stream>>>(Tr, Ti, Dr, Di, (float*)nullptr, (float*)nullptr, Gp, 1);

    // 4) windowed gather at the 262144 nonuniform points
    nfft_gather_kernel<<<MPTS / 256, 256, 0, stream>>>(x, Gp, out);
}